// Trainer_58909771432011
// MI455X (gfx1250) — compile-verified
//
#include <hip/hip_runtime.h>

// Detection loss (G=14 grid, C=20 classes, CH=51 channels), fp32.
// Single 256-thread block (8 wave32s): per-thread masked sum-of-squares over
// the 784 (d,i,j) cells, then a WMMA_F32_16X16X4_F32 all-ones-B reduction
// collapses 256 threads x 12 accumulators into 12 totals in one matrix op.

#define GRID 14
#define NCLS 20
#define NCH  51   // 3 + NCLS + 2*GRID
#define NBOX 16

typedef float v2f __attribute__((ext_vector_type(2)));
typedef float v8f __attribute__((ext_vector_type(8)));

__global__ __launch_bounds__(256) void Trainer_58909771432011_kernel(
    const float* __restrict__ o,       // (4, G, G, 4, CH)
    const float* __restrict__ bboxes,  // (1, 16, 4) = x0,y0,x1,y1
    const int*   __restrict__ labels,  // (1, 16)
    const int*   __restrict__ Wptr,    // scalar W
    float*       __restrict__ out)     // 12 floats
{
    __shared__ float s_acc[16][256];   // component-major partials (12 used, 4 zero pad)
    __shared__ int   s_psx[NBOX * 4];
    __shared__ int   s_psy[NBOX * 4];
    __shared__ float s_pdx[NBOX * 4];
    __shared__ float s_pdy[NBOX * 4];
    __shared__ int   s_cx[NBOX];
    __shared__ int   s_cy[NBOX];
    __shared__ float s_cdx[NBOX];
    __shared__ float s_cdy[NBOX];
    __shared__ int   s_lab[NBOX];
    __shared__ float s_final[12];

    const int tid = threadIdx.x;

    // ---- Phase 0: ground-truth conversion (one thread per box) ----
    if (tid < NBOX) {
        const float s = (float)GRID / (float)Wptr[0];
        const float x0 = bboxes[tid * 4 + 0] * s;
        const float y0 = bboxes[tid * 4 + 1] * s;
        const float x1 = bboxes[tid * 4 + 2] * s;
        const float y1 = bboxes[tid * 4 + 3] * s;
        const float cxf[4] = {x0, x1, x0, x1};
        const float cyf[4] = {y0, y0, y1, y1};
        for (int k = 0; k < 4; ++k) {
            const float fx = floorf(cxf[k]);
            const float fy = floorf(cyf[k]);
            s_psx[tid * 4 + k] = (int)fx;
            s_psy[tid * 4 + k] = (int)fy;
            s_pdx[tid * 4 + k] = cxf[k] - fx;
            s_pdy[tid * 4 + k] = cyf[k] - fy;
        }
        const float ctx = (x0 + x1) * 0.5f;
        const float cty = (y0 + y1) * 0.5f;
        const float fcx = floorf(ctx);
        const float fcy = floorf(cty);
        s_cx[tid] = (int)fcx;
        s_cy[tid] = (int)fcy;
        s_cdx[tid] = ctx - fcx;
        s_cdy[tid] = cty - fcy;
        s_lab[tid] = labels[tid];
    }
    __syncthreads();

    // acc: 0 loss1_p, 1 nopt_p, 2 loss2_p(raw), 3 loss3_p(raw), 4 loss4_p,
    //      5 loss1_c, 6 loss2_c(raw), 7 loss3_c(raw), 8 loss4_c,
    //      9 center_pt(raw), 10 center_nopt(raw), 11 center_off(raw)
    float acc[12];
    #pragma unroll
    for (int i = 0; i < 12; ++i) acc[i] = 0.0f;

    // ---- Phase 1: masked squared-error accumulation over 4*14*14 cells ----
    for (int cell = tid; cell < 4 * GRID * GRID; cell += 256) {
        const int d   = cell / (GRID * GRID);
        const int rem = cell - d * (GRID * GRID);
        const int ga  = rem / GRID;       // x-grid index
        const int gb  = rem - ga * GRID;  // y-grid index

        bool mp = false, mc = false, fp = false, fc = false;
        int  wp = 0, wc = 0;
        for (int n = 0; n < NBOX; ++n) {
            const bool ex = (s_psx[n * 4 + d] == ga);
            const bool ey = (s_psy[n * 4 + d] == gb);
            if (ex && ey) mp = true;
            if (!fp && (ex || ey)) { wp = n; fp = true; }
            const bool ecx = (s_cx[n] == ga);
            const bool ecy = (s_cy[n] == gb);
            if (ecx && ecy) mc = true;
            if (!fc && (ecx || ecy)) { wc = n; fc = true; }
        }

        const float* cellp = o + (size_t)cell * 4 * NCH;

        // --- corner ("p") terms over dd = 0,1 ---
        if (mp) {
            const int   lab = s_lab[wp];
            const float tx  = s_pdx[wp * 4 + d];
            const float ty  = s_pdy[wp * 4 + d];
            const int   lcx = s_cx[wp];
            const int   lcy = s_cy[wp];
            #pragma unroll
            for (int dd = 0; dd < 2; ++dd) {
                const float* p = cellp + dd * NCH;
                const float e = p[0] - 1.0f;
                acc[0] += e * e;
                for (int c = 0; c < NCLS; ++c) {
                    const float t = (c == lab) ? 1.0f : 0.0f;
                    const float v = p[1 + c] - t;
                    acc[2] += v * v;
                }
                const float v0 = p[1 + NCLS] - tx;
                const float v1 = p[2 + NCLS] - ty;
                acc[3] += v0 * v0 + v1 * v1;
                for (int g = 0; g < GRID; ++g) {
                    const float gx = (g == lcx) ? 1.0f : 0.0f;
                    const float gy = (g == lcy) ? 1.0f : 0.0f;
                    const float vx = p[3 + NCLS + g] - gx;
                    const float vy = p[3 + NCLS + GRID + g] - gy;
                    acc[4] += vx * vx + vy * vy;
                }
            }
        } else {
            #pragma unroll
            for (int dd = 0; dd < 2; ++dd) {
                const float e = cellp[dd * NCH];
                acc[1] += e * e;
            }
        }

        // --- center ("c") terms over dd = 2,3 ---
        {
            const float e2 = cellp[2 * NCH];  // o[d,ga,gb,2,0]
            if (mc) {
                const int   lab = s_lab[wc];
                const float tx  = s_cdx[wc];
                const float ty  = s_cdy[wc];
                const int   lpx = s_psx[wc * 4 + d];
                const int   lcy = s_cy[wc];
                #pragma unroll
                for (int dd = 2; dd < 4; ++dd) {
                    const float* p = cellp + dd * NCH;
                    const float e = p[0] - 1.0f;
                    acc[5] += e * e;
                    for (int c = 0; c < NCLS; ++c) {
                        const float t = (c == lab) ? 1.0f : 0.0f;
                        const float v = p[1 + c] - t;
                        acc[6] += v * v;
                    }
                    const float v0 = p[1 + NCLS] - tx;
                    const float v1 = p[2 + NCLS] - ty;
                    acc[7] += v0 * v0 + v1 * v1;
                    for (int g = 0; g < GRID; ++g) {
                        const float gx = (g == lpx) ? 1.0f : 0.0f;
                        const float gy = (g == lcy) ? 1.0f : 0.0f;
                        const float vx = p[3 + NCLS + g] - gx;
                        const float vy = p[3 + NCLS + GRID + g] - gy;
                        acc[8] += vx * vx + vy * vy;
                    }
                }
                const float ee = e2 - 1.0f;
                acc[9] += ee * ee;
                const float o0 = cellp[2 * NCH + 1 + NCLS] - tx;
                const float o1 = cellp[2 * NCH + 2 + NCLS] - ty;
                acc[11] += o0 * o0 + o1 * o1;
            } else {
                acc[10] += e2 * e2;
            }
        }
    }

    // ---- Phase 2: spill partials, component-major ----
    #pragma unroll
    for (int i = 0; i < 12; ++i) s_acc[i][tid] = acc[i];
    #pragma unroll
    for (int i = 12; i < 16; ++i) s_acc[i][tid] = 0.0f;
    __syncthreads();

    // ---- Phase 3: WMMA reduction (wave 0, EXEC all-ones) ----
    // A(16x4): row m = component m; K-slots = four 64-thread quarters.
    //   Lane L (L<16): VGPR0=K0, VGPR1=K1; lane L>=16: VGPR0=K2, VGPR1=K3.
    // B = all-ones (layout independent). D[m][n] = total sum of component m.
    if (tid < 32) {
        const int comp = tid & 15;
        const int half = tid >> 4;
        float a0 = 0.0f, a1 = 0.0f;
        const int base0 = half * 128;
        for (int t = 0; t < 64; ++t) {
            a0 += s_acc[comp][base0 + t];
            a1 += s_acc[comp][base0 + 64 + t];
        }
        v2f A; A[0] = a0;   A[1] = a1;
        v2f B; B[0] = 1.0f; B[1] = 1.0f;
        v8f Cacc = {};
        v8f D = __builtin_amdgcn_wmma_f32_16x16x4_f32(
            false, A, false, B, (short)0, Cacc, false, false);
        // D layout: VGPR r, lanes 0-15 -> D[r][lane]; lanes 16-31 -> D[r+8][lane-16]
        if (tid == 0) {
            #pragma unroll
            for (int m = 0; m < 8; ++m) s_final[m] = D[m];
        } else if (tid == 16) {
            #pragma unroll
            for (int m = 0; m < 4; ++m) s_final[8 + m] = D[m];
        }
    }
    __syncthreads();

    // ---- Phase 4: apply weights, emit the 12 outputs ----
    if (tid == 0) {
        const float l1p = s_final[0];
        const float np  = s_final[1];
        const float l2p = s_final[2];          // W_CLASS = 1
        const float l3p = 5.0f * s_final[3];   // W_COORD = 5
        const float l4p = s_final[4];          // W_LINK = 1
        const float l1c = s_final[5];
        const float l2c = s_final[6];
        const float l3c = 5.0f * s_final[7];
        const float l4c = s_final[8];
        const float cpt = 2.0f  * s_final[9];
        const float cnp = 2.0f  * s_final[10];
        const float cof = 10.0f * s_final[11]; // 2 * W_COORD

        const float loss1 = l1p + l1c;
        const float loss2 = l2p + l2c;
        const float loss3 = l3p + l3c;
        const float loss4 = l4p + l4c;
        const float loss_pt   = loss1 + loss2 + loss3 + loss4;
        const float loss_nopt = np + cnp;

        out[0]  = loss1;
        out[1]  = loss2;
        out[2]  = loss3;
        out[3]  = loss4;
        out[4]  = loss_pt;
        out[5]  = loss_nopt;
        out[6]  = loss1 + 0.05f * loss_nopt;        // add_pexist_nopt
        out[7]  = cpt + cof + 0.05f * cnp;          // center_exist_loss
        out[8]  = cpt;                              // center_pt_loss
        out[9]  = cnp;                              // center_nopt_loss
        out[10] = cof;                              // center_offset_loss
        out[11] = loss_pt + 0.05f * loss_nopt;      // total_loss
    }
}

extern "C" void kernel_launch(void* const* d_in, const int* in_sizes, int n_in,
                              void* d_out, int out_size, void* d_ws, size_t ws_size,
                              hipStream_t stream) {
    (void)in_sizes; (void)n_in; (void)out_size; (void)d_ws; (void)ws_size;
    const float* o      = (const float*)d_in[0];
    const float* bboxes = (const float*)d_in[1];
    const int*   labels = (const int*)d_in[2];
    // d_in[3] = H (unused by reference math), d_in[4] = W
    const int*   Wptr   = (const int*)d_in[4];
    float* out = (float*)d_out;
    Trainer_58909771432011_kernel<<<1, 256, 0, stream>>>(o, bboxes, labels, Wptr, out);
}